// PrefrontalCortex_44727789421230
// MI455X (gfx1250) — compile-verified
//
#include <hip/hip_runtime.h>
#include <cmath>

typedef __attribute__((ext_vector_type(2))) float v2f;
typedef __attribute__((ext_vector_type(8))) float v8f;

#define MEM_SLOTS 4096
#define MEM_DIM   1024
#define HID       2048

__device__ __forceinline__ float sigmoidf_(float x) { return 1.0f / (1.0f + expf(-x)); }

// ---------------------------------------------------------------------------
// Generic WMMA-f32 GEMV:  out[r] = act( dot(W1[r,:],v1) [+ dot(W2[r,:],v2)]
//                                        [+ b1[r]] [+ b2[r]] )
// One wave -> 16 output rows.  Uses V_WMMA_F32_16X16X4_F32 with B = vector
// replicated across all 16 columns; accumulator column N=0 read back.
// A layout (ISA 7.12.2): VGPR0 = K{0|2} lanes{0-15|16-31}, VGPR1 = K{1|3}.
// ---------------------------------------------------------------------------
__global__ __launch_bounds__(256) void gemv_wmma_kernel(
    const float* __restrict__ W1, const float* __restrict__ v1, int K1,
    const float* __restrict__ W2, const float* __restrict__ v2, int K2,
    const float* __restrict__ b1, const float* __restrict__ b2,
    float* __restrict__ out, int rows, int act)
{
    const int lane     = threadIdx.x & 31;
    const int wave     = threadIdx.x >> 5;
    const int row_base = (blockIdx.x * 8 + wave) * 16;   // wave-uniform
    if (row_base >= rows) return;                        // whole-wave exit only

    int r = row_base + (lane & 15);
    if (r >= rows) r = rows - 1;                         // clamp loads; stores guarded
    const int koff = (lane >> 4) << 1;                   // 0 for lanes 0-15, 2 for 16-31

    v8f c = {0.f, 0.f, 0.f, 0.f, 0.f, 0.f, 0.f, 0.f};

    {
        const float* wrow = W1 + (size_t)r * K1;
        for (int k = 0; k < K1; k += 4) {
            v2f a = *(const v2f*)(wrow + k + koff);      // A[m, k+koff .. k+koff+1]
            v2f b = *(const v2f*)(v1   + k + koff);      // B[k,n] = v[k] for all n
            c = __builtin_amdgcn_wmma_f32_16x16x4_f32(false, a, false, b,
                                                      (short)0, c, false, false);
        }
    }
    if (W2) {
        const float* wrow = W2 + (size_t)r * K2;
        for (int k = 0; k < K2; k += 4) {
            v2f a = *(const v2f*)(wrow + k + koff);
            v2f b = *(const v2f*)(v2   + k + koff);
            c = __builtin_amdgcn_wmma_f32_16x16x4_f32(false, a, false, b,
                                                      (short)0, c, false, false);
        }
    }

    // C/D layout: lane 0 holds D[0..7, N=0] in c[0..7]; lane 16 holds D[8..15, N=0].
    if ((lane & 15) == 0) {
        const int mbase = row_base + (lane >> 4) * 8;
        #pragma unroll
        for (int j = 0; j < 8; ++j) {
            const int rr = mbase + j;
            if (rr < rows) {
                float val = c[j];
                if (b1) val += b1[rr];
                if (b2) val += b2[rr];
                if (act == 1) val = tanhf(val);
                out[rr] = val;
            }
        }
    }
}

// softmax over sim[4096] -> read_weights; usage_new = usage + read_weights
__global__ __launch_bounds__(1024) void softmax_read_kernel(
    const float* __restrict__ simv, const float* __restrict__ usage,
    float* __restrict__ rw, float* __restrict__ usage_out)
{
    __shared__ float red[1024];
    const int t = threadIdx.x;
    float m = -INFINITY;
    #pragma unroll
    for (int i = 0; i < 4; ++i) m = fmaxf(m, simv[t + i * 1024]);
    red[t] = m; __syncthreads();
    for (int s = 512; s > 0; s >>= 1) { if (t < s) red[t] = fmaxf(red[t], red[t + s]); __syncthreads(); }
    const float mx = red[0]; __syncthreads();
    float e[4]; float sum = 0.f;
    #pragma unroll
    for (int i = 0; i < 4; ++i) { e[i] = expf(simv[t + i * 1024] - mx); sum += e[i]; }
    red[t] = sum; __syncthreads();
    for (int s = 512; s > 0; s >>= 1) { if (t < s) red[t] += red[t + s]; __syncthreads(); }
    const float inv = 1.0f / red[0];
    #pragma unroll
    for (int i = 0; i < 4; ++i) {
        const int idx = t + i * 1024;
        const float w = e[i] * inv;
        rw[idx] = w;
        usage_out[idx] = usage[idx] + w;
    }
}

// read_vec partials: block (mx, sy) sums 256 slots for 256 columns (coalesced in m)
__global__ __launch_bounds__(256) void readvec_partial_kernel(
    const float* __restrict__ mem, const float* __restrict__ rw,
    float* __restrict__ part)
{
    const int m  = blockIdx.x * 256 + threadIdx.x;
    const int s0 = blockIdx.y * 256;
    float acc = 0.f;
    for (int s = 0; s < 256; ++s)
        acc += rw[s0 + s] * mem[(size_t)(s0 + s) * MEM_DIM + m];
    part[blockIdx.y * MEM_DIM + m] = acc;
}

__global__ __launch_bounds__(256) void readvec_reduce_kernel(
    const float* __restrict__ part, float* __restrict__ inp_rv,
    float* __restrict__ outc_rv)
{
    const int m = blockIdx.x * 256 + threadIdx.x;
    float acc = 0.f;
    #pragma unroll
    for (int p = 0; p < 16; ++p) acc += part[p * MEM_DIM + m];
    inp_rv[m]  = acc;
    outc_rv[m] = acc;
}

// LSTM pointwise: gates = [i | f | g | o], each 2048
__global__ __launch_bounds__(256) void lstm_kernel(
    const float* __restrict__ gates, const float* __restrict__ c_prev,
    float* __restrict__ h_out, float* __restrict__ c_out,
    float* __restrict__ outc_h)
{
    const int i = blockIdx.x * 256 + threadIdx.x;
    const float ig = sigmoidf_(gates[i]);
    const float fg = sigmoidf_(gates[HID + i]);
    const float gg = tanhf(gates[2 * HID + i]);
    const float og = sigmoidf_(gates[3 * HID + i]);
    const float c  = fg * c_prev[i] + ig * gg;
    const float h  = og * tanhf(c);
    c_out[i] = c;
    h_out[i] = h;
    outc_h[i] = h;
}

// erase/add vectors + softmax(w_logits)*write_gate from params[7169]
__global__ __launch_bounds__(1024) void head_post_kernel(
    const float* __restrict__ params, float* __restrict__ wvec,
    float* __restrict__ erasev, float* __restrict__ addv)
{
    __shared__ float red[1024];
    const int t = threadIdx.x;
    erasev[t] = sigmoidf_(params[MEM_DIM + t]);
    addv[t]   = tanhf(params[2 * MEM_DIM + t]);
    const float* logits = params + 3 * MEM_DIM;
    float m = -INFINITY;
    #pragma unroll
    for (int i = 0; i < 4; ++i) m = fmaxf(m, logits[t + i * 1024]);
    red[t] = m; __syncthreads();
    for (int s = 512; s > 0; s >>= 1) { if (t < s) red[t] = fmaxf(red[t], red[t + s]); __syncthreads(); }
    const float mx = red[0]; __syncthreads();
    float e[4]; float sum = 0.f;
    #pragma unroll
    for (int i = 0; i < 4; ++i) { e[i] = expf(logits[t + i * 1024] - mx); sum += e[i]; }
    red[t] = sum; __syncthreads();
    for (int s = 512; s > 0; s >>= 1) { if (t < s) red[t] += red[t + s]; __syncthreads(); }
    const float scale = sigmoidf_(params[3 * MEM_DIM + MEM_SLOTS]) / red[0];
    #pragma unroll
    for (int i = 0; i < 4; ++i) wvec[t + i * 1024] = e[i] * scale;
}

// new_memory = where(usage<0.1, 0, mem) * (1 - w[s]*erase[m]) + w[s]*add[m]
__global__ __launch_bounds__(256) void memupdate_kernel(
    const float* __restrict__ mem, const float* __restrict__ usage_new,
    const float* __restrict__ w, const float* __restrict__ erasev,
    const float* __restrict__ addv, float* __restrict__ out)
{
    const int s  = blockIdx.x;
    const int m0 = threadIdx.x * 4;
    const float u  = usage_new[s];
    const float ws = w[s];
    float4 mm;
    if (u < 0.1f) { mm.x = mm.y = mm.z = mm.w = 0.f; }
    else mm = *(const float4*)(mem + (size_t)s * MEM_DIM + m0);
    const float4 e = *(const float4*)(erasev + m0);
    const float4 a = *(const float4*)(addv + m0);
    float4 o;
    o.x = mm.x * (1.0f - ws * e.x) + ws * a.x;
    o.y = mm.y * (1.0f - ws * e.y) + ws * a.y;
    o.z = mm.z * (1.0f - ws * e.z) + ws * a.z;
    o.w = mm.w * (1.0f - ws * e.w) + ws * a.w;
    *(float4*)(out + (size_t)s * MEM_DIM + m0) = o;
}

extern "C" void kernel_launch(void* const* d_in, const int* in_sizes, int n_in,
                              void* d_out, int out_size, void* d_ws, size_t ws_size,
                              hipStream_t stream)
{
    (void)in_sizes; (void)n_in; (void)out_size; (void)ws_size;
    const float* x      = (const float*)d_in[0];
    const float* h_prev = (const float*)d_in[1];
    const float* c_prev = (const float*)d_in[2];
    const float* memory = (const float*)d_in[3];
    const float* usage  = (const float*)d_in[4];
    const float* read_W = (const float*)d_in[5];
    const float* read_b = (const float*)d_in[6];
    const float* W_ih   = (const float*)d_in[7];
    const float* b_ih   = (const float*)d_in[8];
    const float* W_hh   = (const float*)d_in[9];
    const float* b_hh   = (const float*)d_in[10];
    const float* head_W = (const float*)d_in[11];
    const float* head_b = (const float*)d_in[12];
    const float* out_W  = (const float*)d_in[13];
    const float* out_b  = (const float*)d_in[14];

    // d_out layout: out(2048) | h_new(2048) | c_new(2048) | new_memory(4096*1024) | usage_new(4096)
    float* out_vec   = (float*)d_out;
    float* h_out     = out_vec + 2048;
    float* c_out     = h_out + 2048;
    float* mem_out   = c_out + 2048;
    float* usage_out = mem_out + (size_t)MEM_SLOTS * MEM_DIM;

    // workspace layout (floats); all offsets 16B-aligned
    float* ws      = (float*)d_ws;
    float* rk      = ws + 0;       // 1024  read_key
    float* simv    = ws + 1024;    // 4096  similarity
    float* rw      = ws + 5120;    // 4096  read_weights
    float* part    = ws + 9216;    // 16384 read_vec partials (16 x 1024)
    float* inpv    = ws + 25600;   // 3072  [x | read_vec]
    float* outc    = ws + 28672;   // 3072  [h_new | read_vec]
    float* gates   = ws + 31744;   // 8192
    float* paramsv = ws + 39936;   // 7169 (padded to 7172)
    float* wvec    = ws + 47108;   // 4096  write weights
    float* erasev  = ws + 51204;   // 1024
    float* addv    = ws + 52228;   // 1024

    // inp[0:2048] = x
    hipMemcpyAsync(inpv, x, 2048 * sizeof(float), hipMemcpyDeviceToDevice, stream);

    // read_key = tanh(read_W @ h_prev + read_b)        (1024 x 2048)
    gemv_wmma_kernel<<<(1024 + 127) / 128, 256, 0, stream>>>(
        read_W, h_prev, 2048, nullptr, nullptr, 0, read_b, nullptr, rk, 1024, 1);

    // sim = memory @ read_key                          (4096 x 1024)
    gemv_wmma_kernel<<<(4096 + 127) / 128, 256, 0, stream>>>(
        memory, rk, 1024, nullptr, nullptr, 0, nullptr, nullptr, simv, 4096, 0);

    // read_weights = softmax(sim); usage_new = usage + read_weights
    softmax_read_kernel<<<1, 1024, 0, stream>>>(simv, usage, rw, usage_out);

    // read_vec = read_weights @ memory (two-stage deterministic reduction)
    readvec_partial_kernel<<<dim3(4, 16), 256, 0, stream>>>(memory, rw, part);
    readvec_reduce_kernel<<<4, 256, 0, stream>>>(part, inpv + 2048, outc + 2048);

    // gates = W_ih @ [x|read_vec] + W_hh @ h_prev + b_ih + b_hh   (8192 rows)
    gemv_wmma_kernel<<<(8192 + 127) / 128, 256, 0, stream>>>(
        W_ih, inpv, 3072, W_hh, h_prev, 2048, b_ih, b_hh, gates, 8192, 0);

    // LSTM pointwise -> h_new, c_new (also h_new into outc[0:2048])
    lstm_kernel<<<8, 256, 0, stream>>>(gates, c_prev, h_out, c_out, outc);

    // params = head_W @ h_new + head_b                 (7169 x 2048)
    gemv_wmma_kernel<<<(7169 + 127) / 128, 256, 0, stream>>>(
        head_W, outc, 2048, nullptr, nullptr, 0, head_b, nullptr, paramsv, 7169, 0);

    // erase / add / w = softmax(logits)*write_gate
    head_post_kernel<<<1, 1024, 0, stream>>>(paramsv, wvec, erasev, addv);

    // new_memory
    memupdate_kernel<<<MEM_SLOTS, 256, 0, stream>>>(
        memory, usage_out, wvec, erasev, addv, mem_out);

    // out = out_W @ [h_new|read_vec] + out_b           (2048 x 3072)
    gemv_wmma_kernel<<<(2048 + 127) / 128, 256, 0, stream>>>(
        out_W, outc, 3072, nullptr, nullptr, 0, out_b, nullptr, out_vec, 2048, 0);
}